// Model_24558622998904
// MI455X (gfx1250) — compile-verified
//
#include <hip/hip_runtime.h>
#include <hip/hip_bf16.h>
#include <math.h>

// Graph-transformer layer for MI455X (gfx1250, wave32, WMMA).
// N=50000 nodes, E=800000 edges, D=128, H=8 heads, HD=16.

#define N_NODES 50000
#define N_EDGES 800000
#define DIM     128
#define NHEAD   8
#define LN_EPS  1e-5f

typedef float v2f __attribute__((ext_vector_type(2)));
typedef float v8f __attribute__((ext_vector_type(8)));

// ---- fp32 WMMA 16x16x4: D = A(16x4) * B(4x16) + C, full fp32 precision ----
__device__ __forceinline__ v8f wmma4(v2f a, v2f b, v8f c) {
  return __builtin_amdgcn_wmma_f32_16x16x4_f32(
      /*neg_a=*/false, a, /*neg_b=*/false, b,
      /*c_mod=*/(short)0, c, /*reuse_a=*/false, /*reuse_b=*/false);
}

// ---- CDNA5 async memory->LDS DMA (ASYNCcnt-tracked), 16B per lane ----
// VDST VGPR holds the LDS byte offset (low 32 bits of the generic LDS addr,
// per the flat-aperture truncation rule); VADDR is the 64-bit global address.
__device__ __forceinline__ void async_g2lds_b128(float* lds_dst,
                                                 const float* gsrc) {
  unsigned ldsoff = (unsigned)(size_t)(void*)lds_dst;
  unsigned long long ga = (unsigned long long)gsrc;
  asm volatile("global_load_async_to_lds_b128 %0, %1, off"
               :
               : "v"(ldsoff), "v"(ga)
               : "memory");
}
__device__ __forceinline__ void wait_async0() {
  asm volatile("s_wait_asynccnt 0" ::: "memory");
}

// order-preserving float <-> uint map for atomicMax-based segment max
__device__ __forceinline__ unsigned fenc(float f) {
  unsigned u = __float_as_uint(f);
  return (u & 0x80000000u) ? ~u : (u | 0x80000000u);
}
__device__ __forceinline__ float fdec(unsigned e) {
  return (e & 0x80000000u) ? __uint_as_float(e & 0x7FFFFFFFu)
                           : __uint_as_float(~e);
}
#define ENC_NEG_INF 0x007FFFFFu   // fenc(-inf)

__device__ __forceinline__ float wred32(float x) {
#pragma unroll
  for (int m = 16; m >= 1; m >>= 1) x += __shfl_xor(x, m, 32);
  return x;
}
__device__ __forceinline__ float dot4(const float4& a, const float4& b) {
  return a.x * b.x + a.y * b.y + a.z * b.z + a.w * b.w;
}

// ===========================================================================
// Generic C[M,128] = epi(A[M,128] @ W[128,128]).  MODE: 0 none, 1 relu, 2 +R
// (compile-time -> branch-free epilogue).  128 threads = 4 waves; 64 rows per
// block; A tile staged in LDS via async DMA (stride 132 floats: 16B-aligned
// rows, bank-spread).  32 x V_WMMA_F32_16X16X4_F32 per 16x16 tile.
// ===========================================================================
template <int MODE>
__global__ __launch_bounds__(128) void gemm128(
    const float* __restrict__ A, const float* __restrict__ W,
    float* __restrict__ C, const float* __restrict__ R, int M) {
  __shared__ float lA[64 * 132];
  const int t = threadIdx.x;
  const int lane = t & 31, wv = t >> 5;
  const int lm = lane & 15, lh = lane >> 4;
  const int r0 = blockIdx.x * 64;
  const bool full = (r0 + 64 <= M);

  __builtin_prefetch(W, 0, 0);  // weights: keep hot (global_prefetch_b8)

  // async stage of the 64x128 A tile (memory -> LDS, no VGPR round-trip)
  if (full) {
#pragma unroll
    for (int i = 0; i < 16; ++i) {
      int lin = i * 128 + t;
      int row = lin >> 5, c4 = lin & 31;
      async_g2lds_b128(&lA[row * 132 + c4 * 4],
                       A + (size_t)(r0 + row) * DIM + c4 * 4);
    }
  } else {
#pragma unroll
    for (int i = 0; i < 16; ++i) {
      int lin = i * 128 + t;
      int row = lin >> 5, c4 = lin & 31;
      int gr = r0 + row;
      if (gr > M - 1) gr = M - 1;   // clamp tail rows (values unused)
      async_g2lds_b128(&lA[row * 132 + c4 * 4],
                       A + (size_t)gr * DIM + c4 * 4);
    }
  }
  wait_async0();
  __syncthreads();

  const float* lrow = &lA[(wv * 16 + lm) * 132];
#pragma unroll 1
  for (int ct = 0; ct < 8; ++ct) {
    v8f acc = {0.f, 0.f, 0.f, 0.f, 0.f, 0.f, 0.f, 0.f};
    const int n = ct * 16 + lm;
#pragma unroll 8
    for (int kk = 0; kk < 32; ++kk) {
      const int kb = kk * 4 + 2 * lh;  // A frag: lanes 0-15 K,K+1 ; 16-31 K+2,K+3
      v2f a = *(const v2f*)(lrow + kb);
      v2f b;                           // B frag: VGPR0 rows K/K+2, VGPR1 rows K+1/K+3
      b.x = W[(size_t)kb * DIM + n];
      b.y = W[(size_t)(kb + 1) * DIM + n];
      acc = wmma4(a, b, acc);
    }
    const int rbase = r0 + wv * 16 + 8 * lh;
    if (full) {                        // common path: straight-line stores
#pragma unroll
      for (int r = 0; r < 8; ++r) {
        float val = acc[r];
        if (MODE == 1) val = fmaxf(val, 0.f);
        if (MODE == 2) val += R[(size_t)(rbase + r) * DIM + n];
        C[(size_t)(rbase + r) * DIM + n] = val;
      }
    } else {
#pragma unroll
      for (int r = 0; r < 8; ++r) {
        if (rbase + r < M) {
          float val = acc[r];
          if (MODE == 1) val = fmaxf(val, 0.f);
          if (MODE == 2) val += R[(size_t)(rbase + r) * DIM + n];
          C[(size_t)(rbase + r) * DIM + n] = val;
        }
      }
    }
  }
}

// ===========================================================================
// Edge pass 1: per-edge [bias|gate] via WMMA against [We|Wg] (128x16), QK
// head-dots via VALU, then logits/gates stores + atomicMax segment max.
// 128 threads = 4 waves; 64 edges/block (E % 64 == 0).
// ===========================================================================
__global__ __launch_bounds__(128) void edge_logits(
    const float* __restrict__ ef, const int* __restrict__ src,
    const int* __restrict__ dst, const float* __restrict__ q,
    const float* __restrict__ kf, const float* __restrict__ We,
    const float* __restrict__ Wg, float* __restrict__ logits,
    float* __restrict__ gates, unsigned* __restrict__ mkey) {
  __shared__ float lE[64 * 132];
  __shared__ float tb[4][16][17];   // per-wave 16 edges x (8 bias | 8 gate logit)
  const int t = threadIdx.x;
  const int lane = t & 31, wv = t >> 5;
  const int lm = lane & 15, lh = lane >> 4;
  const int e0 = blockIdx.x * 64;

  __builtin_prefetch(src + e0, 0, 0);
  __builtin_prefetch(dst + e0, 0, 0);

  // async stage of the 64x128 edge_feat tile (streamed once from HBM)
#pragma unroll
  for (int i = 0; i < 16; ++i) {
    int lin = i * 128 + t;
    int row = lin >> 5, c4 = lin & 31;
    async_g2lds_b128(&lE[row * 132 + c4 * 4],
                     ef + (size_t)(e0 + row) * DIM + c4 * 4);
  }
  wait_async0();
  __syncthreads();

  // T(16x16) = edge_tile(16x128) @ [We | Wg](128x16)
  const float* lrow = &lE[(wv * 16 + lm) * 132];
  const float* Wsel = (lm < 8) ? We : Wg;   // virtual concat along N
  const int nn = lm & 7;
  v8f acc = {0.f, 0.f, 0.f, 0.f, 0.f, 0.f, 0.f, 0.f};
#pragma unroll 8
  for (int kk = 0; kk < 32; ++kk) {
    const int kb = kk * 4 + 2 * lh;
    v2f a = *(const v2f*)(lrow + kb);
    v2f b;
    b.x = Wsel[(size_t)kb * NHEAD + nn];
    b.y = Wsel[(size_t)(kb + 1) * NHEAD + nn];
    acc = wmma4(a, b, acc);
  }
#pragma unroll
  for (int r = 0; r < 8; ++r) tb[wv][r + 8 * lh][lm] = acc[r];
  __syncthreads();

  // per-lane: edge ei = lm, heads [4*lh .. 4*lh+3]
  const int ei = lm;
  const int e = e0 + wv * 16 + ei;
  const int s = src[e], d = dst[e];
  const float4* qp = (const float4*)(q + (size_t)s * DIM);
  const float4* kp = (const float4*)(kf + (size_t)d * DIM);
#pragma unroll
  for (int j = 0; j < 4; ++j) {
    const int h = lh * 4 + j;
    float a = 0.f;
#pragma unroll
    for (int c = 0; c < 4; ++c) {
      float4 qv = qp[h * 4 + c], kv = kp[h * 4 + c];
      a += dot4(qv, kv);
    }
    a = fminf(5.f, fmaxf(-5.f, a));
    const float logit = (a + tb[wv][ei][h]) * 4.0f;      // * sqrt(HD)
    const float g = 1.f / (1.f + expf(-tb[wv][ei][8 + h]));
    logits[(size_t)e * NHEAD + h] = logit;
    gates[(size_t)e * NHEAD + h] = g;
    atomicMax(mkey + (size_t)d * NHEAD + h, fenc(logit));
  }
}

// Edge pass 2: p = exp(logit - m[dst]); denom += p; gates <- p*gate (in place)
__global__ __launch_bounds__(256) void edge_softmax_p(
    const int* __restrict__ dst, const float* __restrict__ logits,
    float* __restrict__ gates_pg, const unsigned* __restrict__ mkey,
    float* __restrict__ denom) {
  int tid = blockIdx.x * blockDim.x + threadIdx.x;
  if (tid >= N_EDGES * 2) return;
  const int e = tid >> 1, half = tid & 1;
  const int d = dst[e];
  const size_t lo = (size_t)e * NHEAD + half * 4;
  const size_t no = (size_t)d * NHEAD + half * 4;
  float4 lg = *(const float4*)(logits + lo);
  float4 gt = *(const float4*)(gates_pg + lo);
  float4 p;
  p.x = expf(lg.x - fdec(mkey[no + 0]));
  p.y = expf(lg.y - fdec(mkey[no + 1]));
  p.z = expf(lg.z - fdec(mkey[no + 2]));
  p.w = expf(lg.w - fdec(mkey[no + 3]));
  atomicAdd(denom + no + 0, p.x);
  atomicAdd(denom + no + 1, p.y);
  atomicAdd(denom + no + 2, p.z);
  atomicAdd(denom + no + 3, p.w);
  float4 pg; pg.x = p.x * gt.x; pg.y = p.y * gt.y; pg.z = p.z * gt.z; pg.w = p.w * gt.w;
  *(float4*)(gates_pg + lo) = pg;
}

// Edge pass 3: agg[dst] += v[src] * (p*gate)/denom ; 32 threads per edge
__global__ __launch_bounds__(256) void edge_aggregate(
    const int* __restrict__ src, const int* __restrict__ dst,
    const float* __restrict__ v, const float* __restrict__ pg,
    const float* __restrict__ denom, float* __restrict__ agg) {
  int tid = blockIdx.x * blockDim.x + threadIdx.x;
  if (tid >= N_EDGES * 32) return;
  const int e = tid >> 5, c = tid & 31;
  const int h = c >> 2;
  const int d = dst[e], s = src[e];
  float den = denom[(size_t)d * NHEAD + h];
  if (den == 0.f) den = 1.f;
  const float coef = pg[(size_t)e * NHEAD + h] / den;
  float4 vv = *(const float4*)(v + (size_t)s * DIM + c * 4);
  float* ap = agg + (size_t)d * DIM + c * 4;
  atomicAdd(ap + 0, vv.x * coef);
  atomicAdd(ap + 1, vv.y * coef);
  atomicAdd(ap + 2, vv.z * coef);
  atomicAdd(ap + 3, vv.w * coef);
}

// Gated residual + LN1 -> h ; LN2(h) -> ln2h.  Wave per node (wave32).
__global__ __launch_bounds__(128) void gate_ln(
    const float* __restrict__ rst, const float* __restrict__ skip,
    const float* __restrict__ Wgate, const float* __restrict__ g1,
    const float* __restrict__ b1, const float* __restrict__ g2,
    const float* __restrict__ b2, float* __restrict__ hout,
    float* __restrict__ ln2h, int n) {
  const int lane = threadIdx.x & 31, wv = threadIdx.x >> 5;
  const int node = blockIdx.x * 4 + wv;
  if (node >= n) return;
  const int c = lane * 4;
  float4 r4 = *(const float4*)(rst + (size_t)node * DIM + c);
  float4 s4 = *(const float4*)(skip + (size_t)node * DIM + c);
  float4 w0 = *(const float4*)(Wgate + c);
  float4 w1 = *(const float4*)(Wgate + DIM + c);
  float4 w2 = *(const float4*)(Wgate + 2 * DIM + c);
  float part = dot4(r4, w0) + dot4(s4, w1) + dot4(r4, w2) - dot4(s4, w2);
  const float g = 1.f / (1.f + expf(-wred32(part)));
  float4 x;
  x.x = r4.x * g + s4.x * (1.f - g);
  x.y = r4.y * g + s4.y * (1.f - g);
  x.z = r4.z * g + s4.z * (1.f - g);
  x.w = r4.w * g + s4.w * (1.f - g);
  const float mean = wred32(x.x + x.y + x.z + x.w) * (1.f / DIM);
  const float var =
      wred32(x.x * x.x + x.y * x.y + x.z * x.z + x.w * x.w) * (1.f / DIM) -
      mean * mean;
  const float is = rsqrtf(var + LN_EPS);
  float4 ga = *(const float4*)(g1 + c), ba = *(const float4*)(b1 + c);
  float4 hv;
  hv.x = (x.x - mean) * is * ga.x + ba.x;
  hv.y = (x.y - mean) * is * ga.y + ba.y;
  hv.z = (x.z - mean) * is * ga.z + ba.z;
  hv.w = (x.w - mean) * is * ga.w + ba.w;
  *(float4*)(hout + (size_t)node * DIM + c) = hv;
  const float mean2 = wred32(hv.x + hv.y + hv.z + hv.w) * (1.f / DIM);
  const float var2 =
      wred32(hv.x * hv.x + hv.y * hv.y + hv.z * hv.z + hv.w * hv.w) *
          (1.f / DIM) -
      mean2 * mean2;
  const float is2 = rsqrtf(var2 + LN_EPS);
  float4 gb = *(const float4*)(g2 + c), bb = *(const float4*)(b2 + c);
  float4 yv;
  yv.x = (hv.x - mean2) * is2 * gb.x + bb.x;
  yv.y = (hv.y - mean2) * is2 * gb.y + bb.y;
  yv.z = (hv.z - mean2) * is2 * gb.z + bb.z;
  yv.w = (hv.w - mean2) * is2 * gb.w + bb.w;
  *(float4*)(ln2h + (size_t)node * DIM + c) = yv;
}

__global__ __launch_bounds__(256) void init_nh(unsigned* __restrict__ mkey,
                                               float* __restrict__ denom,
                                               int n) {
  int i = blockIdx.x * blockDim.x + threadIdx.x;
  if (i < n) { mkey[i] = ENC_NEG_INF; denom[i] = 0.f; }
}
__global__ __launch_bounds__(256) void zero_f4(float* __restrict__ p, int n4) {
  int i = blockIdx.x * blockDim.x + threadIdx.x;
  if (i < n4) ((float4*)p)[i] = make_float4(0.f, 0.f, 0.f, 0.f);
}

// ===========================================================================
extern "C" void kernel_launch(void* const* d_in, const int* in_sizes, int n_in,
                              void* d_out, int out_size, void* d_ws,
                              size_t ws_size, hipStream_t stream) {
  const float* feat  = (const float*)d_in[0];
  const float* efeat = (const float*)d_in[1];
  const int*   src   = (const int*)d_in[2];
  const int*   dst   = (const int*)d_in[3];
  const float* Wq    = (const float*)d_in[4];
  const float* Wk    = (const float*)d_in[5];
  const float* Wv    = (const float*)d_in[6];
  const float* Wo    = (const float*)d_in[7];
  const float* We    = (const float*)d_in[8];
  const float* Wg    = (const float*)d_in[9];
  const float* Wskip = (const float*)d_in[10];
  const float* Wgate = (const float*)d_in[11];
  const float* ln1g  = (const float*)d_in[12];
  const float* ln1b  = (const float*)d_in[13];
  const float* ln2g  = (const float*)d_in[14];
  const float* ln2b  = (const float*)d_in[15];
  const float* W1    = (const float*)d_in[16];
  const float* W2    = (const float*)d_in[17];

  const int n = N_NODES;
  const size_t ND = (size_t)n * DIM;
  const size_t EH = (size_t)N_EDGES * NHEAD;

  // workspace layout (floats); buffers aliased along the dataflow
  float* q      = (float*)d_ws;          // later: agg, then h
  float* kf     = q + ND;                // later: rst
  float* v      = kf + ND;               // later: ln2h
  float* skip   = v + ND;                // later: t (FFN hidden)
  float* logits = skip + ND;
  float* gates  = logits + EH;           // becomes p*gate after pass 2
  float* denom  = gates + EH;
  unsigned* mkey = (unsigned*)(denom + (size_t)n * NHEAD);
  float* agg  = q;
  float* rst  = kf;
  float* hbuf = q;      // alias agg: agg dead once rst is produced
  float* l2h  = v;      // alias v: dead after aggregation
  float* tbuf = skip;   // alias skip: dead after gate_ln

  const int gemmBlocks = (n + 63) / 64;

  // node projections: q/k/v/skip
  gemm128<0><<<gemmBlocks, 128, 0, stream>>>(feat, Wq,    q,    nullptr, n);
  gemm128<0><<<gemmBlocks, 128, 0, stream>>>(feat, Wk,    kf,   nullptr, n);
  gemm128<0><<<gemmBlocks, 128, 0, stream>>>(feat, Wv,    v,    nullptr, n);
  gemm128<0><<<gemmBlocks, 128, 0, stream>>>(feat, Wskip, skip, nullptr, n);

  // segment-max/denominator init
  init_nh<<<(n * NHEAD + 255) / 256, 256, 0, stream>>>(mkey, denom, n * NHEAD);

  // edge passes
  edge_logits<<<N_EDGES / 64, 128, 0, stream>>>(efeat, src, dst, q, kf, We, Wg,
                                                logits, gates, mkey);
  zero_f4<<<(int)((ND / 4 + 255) / 256), 256, 0, stream>>>(agg, (int)(ND / 4));
  edge_softmax_p<<<(N_EDGES * 2 + 255) / 256, 256, 0, stream>>>(
      dst, logits, gates, mkey, denom);
  edge_aggregate<<<(N_EDGES * 32 + 255) / 256, 256, 0, stream>>>(
      src, dst, v, gates, denom, agg);

  // output projection, gated residual + LNs, FFN
  gemm128<0><<<gemmBlocks, 128, 0, stream>>>(agg, Wo, rst, nullptr, n);
  gate_ln<<<(n + 3) / 4, 128, 0, stream>>>(rst, skip, Wgate, ln1g, ln1b, ln2g,
                                           ln2b, hbuf, l2h, n);
  gemm128<1><<<gemmBlocks, 128, 0, stream>>>(l2h, W1, tbuf, nullptr, n);  // relu
  gemm128<2><<<gemmBlocks, 128, 0, stream>>>(tbuf, W2, (float*)d_out, hbuf, n);
}